// DGLDigitCapsuleLayer_18940805775945
// MI455X (gfx1250) — compile-verified
//
#include <hip/hip_runtime.h>

#define IN_NODES 1152
#define OUT_NODES 10
#define DIM 64          // IN_DIM == OUT_DIM
#define BATCH 64
#define ROUTING 3

typedef __attribute__((ext_vector_type(2))) float v2f;
typedef __attribute__((ext_vector_type(8))) float v8f;

// ---------------------------------------------------------------------------
// x[b, d, i]  ->  xT[i, b, d]   (LDS-tiled transpose, both sides coalesced)
// grid (36, 2, 64)  block (32, 8)
// ---------------------------------------------------------------------------
__global__ void k_transpose_x(const float* __restrict__ x, float* __restrict__ xT) {
    __shared__ float t[32][33];
    const int i0 = blockIdx.x * 32;
    const int d0 = blockIdx.y * 32;
    const int b  = blockIdx.z;
    const int tx = threadIdx.x, ty = threadIdx.y;
    const float* xb = x + (size_t)b * DIM * IN_NODES;
#pragma unroll
    for (int q = 0; q < 4; ++q) {
        int dl = ty + 8 * q;
        t[dl][tx] = xb[(size_t)(d0 + dl) * IN_NODES + i0 + tx];
    }
    __syncthreads();
#pragma unroll
    for (int q = 0; q < 4; ++q) {
        int il = ty + 8 * q;
        xT[((size_t)(i0 + il) * BATCH + b) * DIM + d0 + tx] = t[tx][il];
    }
}

// ---------------------------------------------------------------------------
// u_hat[i,j,b,o] = sum_d xT[i,b,d] * W[i,j,o,d]
// One workgroup (8 wave32) per i. X_i staged once in LDS; W_ij streamed.
// Each wave: two 16x16 tiles, K=64 via 16 steps of v_wmma_f32_16x16x4_f32.
// ---------------------------------------------------------------------------
__global__ void __launch_bounds__(256)
k_uhat(const float* __restrict__ xT, const float* __restrict__ w,
       float* __restrict__ uhat) {
    __shared__ float Xs[64][68];   // pad 68: float4-aligned + 4-bank row rotation
    __shared__ float Ws[64][68];

    const int i    = blockIdx.x;
    const int tid  = threadIdx.x;
    const int wave = tid >> 5;
    const int lane = tid & 31;
    const int half = lane >> 4;    // 0: lanes 0-15, 1: lanes 16-31
    const int l    = lane & 15;
    const int bm   = wave >> 1;            // batch tile 0..3
    const int bn0  = (wave & 1) * 2;       // first of two o-tiles

    // stage X_i (4096 floats)
    {
        const float4* src = (const float4*)(xT + (size_t)i * 4096);
#pragma unroll
        for (int q = 0; q < 4; ++q) {
            int e = tid + q * 256;             // float4 index 0..1023
            float4 val = src[e];
            int row = e >> 4;
            int col = (e & 15) << 2;
            *(float4*)&Xs[row][col] = val;
        }
    }

    const float* wbase = w + (size_t)i * OUT_NODES * 4096;
    for (int j = 0; j < OUT_NODES; ++j) {
        __syncthreads();                       // Ws reuse barrier
        const float4* wsrc = (const float4*)(wbase + (size_t)j * 4096);
        if (j + 1 < OUT_NODES)                 // warm L2/L0 for next W_ij
            __builtin_prefetch(wbase + (size_t)(j + 1) * 4096 + tid * 16, 0, 0);
#pragma unroll
        for (int q = 0; q < 4; ++q) {
            int e = tid + q * 256;
            float4 val = wsrc[e];
            int row = e >> 4;
            int col = (e & 15) << 2;
            *(float4*)&Ws[row][col] = val;
        }
        __syncthreads();

        v8f acc0 = {0.f, 0.f, 0.f, 0.f, 0.f, 0.f, 0.f, 0.f};
        v8f acc1 = {0.f, 0.f, 0.f, 0.f, 0.f, 0.f, 0.f, 0.f};
        const int arow = bm * 16 + l;
        const int n0 = bn0 * 16 + l;
        const int n1 = n0 + 16;
#pragma unroll
        for (int k0 = 0; k0 < 64; k0 += 4) {
            const int kb = k0 + half * 2;
            // A 16x4 fragment: v0/v1 hold K = kb / kb+1 for this half-wave
            v2f a;  a.x = Xs[arow][kb]; a.y = Xs[arow][kb + 1];
            // B 4x16 fragment: B[k][n] = W[o=n][d=k]
            v2f b0; b0.x = Ws[n0][kb]; b0.y = Ws[n0][kb + 1];
            v2f b1; b1.x = Ws[n1][kb]; b1.y = Ws[n1][kb + 1];
            acc0 = __builtin_amdgcn_wmma_f32_16x16x4_f32(
                false, a, false, b0, (short)0, acc0, false, false);
            acc1 = __builtin_amdgcn_wmma_f32_16x16x4_f32(
                false, a, false, b1, (short)0, acc1, false, false);
        }

        float* up = uhat + ((size_t)(i * OUT_NODES + j) << 12);
#pragma unroll
        for (int r = 0; r < 8; ++r) {
            const int m = bm * 16 + half * 8 + r;  // C/D: vgpr r -> rows r, r+8
            up[m * 64 + n0] = acc0[r];
            up[m * 64 + n1] = acc1[r];
        }
    }
}

// ---------------------------------------------------------------------------
__global__ void k_zero(float* __restrict__ p, int n) {
    int t = blockIdx.x * blockDim.x + threadIdx.x;
    if (t < n) p[t] = 0.f;
}

// c[i,:] = softmax(b[i,:]) over j
__global__ void k_softmax(const float* __restrict__ b, float* __restrict__ c) {
    int i = blockIdx.x * blockDim.x + threadIdx.x;
    if (i >= IN_NODES) return;
    const float* br = b + i * OUT_NODES;
    float m = br[0];
#pragma unroll
    for (int j = 1; j < OUT_NODES; ++j) m = fmaxf(m, br[j]);
    float e[OUT_NODES], sum = 0.f;
#pragma unroll
    for (int j = 0; j < OUT_NODES; ++j) { e[j] = __expf(br[j] - m); sum += e[j]; }
    const float inv = 1.0f / sum;
    float* cr = c + i * OUT_NODES;
#pragma unroll
    for (int j = 0; j < OUT_NODES; ++j) cr[j] = e[j] * inv;
}

// s[j,b,o] = sum_i c[i,j] * u_hat[i,j,b,o]     grid(16,10) block 256
__global__ void __launch_bounds__(256)
k_wsum(const float* __restrict__ uhat, const float* __restrict__ c,
       float* __restrict__ s) {
    const int j  = blockIdx.y;
    const int bo = blockIdx.x * 256 + threadIdx.x;     // 0..4095
    const float* up = uhat + (size_t)j * 4096 + bo;
    float acc = 0.f;
#pragma unroll 4
    for (int i = 0; i < IN_NODES; ++i)
        acc += c[i * OUT_NODES + j] * up[(size_t)i * (OUT_NODES * 4096)];
    s[j * 4096 + bo] = acc;
}

// v[j,b,:] = squash(s[j,b,:])    grid 640, block 32 (one wave per row)
__global__ void k_squash(const float* __restrict__ s, float* __restrict__ v) {
    const int row  = blockIdx.x;          // j*64 + b
    const int lane = threadIdx.x;         // 0..31
    const float* sr = s + row * 64;
    const float a0 = sr[lane], a1 = sr[lane + 32];
    float sq = a0 * a0 + a1 * a1;
#pragma unroll
    for (int m = 16; m > 0; m >>= 1) sq += __shfl_xor(sq, m, 32);
    const float scale = (sq / (1.0f + sq)) * rsqrtf(sq + 1e-9f);
    float* vr = v + row * 64;
    vr[lane]      = a0 * scale;
    vr[lane + 32] = a1 * scale;
}

// b[i,j] += (1/B) * sum_{b,o} u_hat[i,j,b,o]*v[j,b,o]   grid(1152,10) block 256
__global__ void __launch_bounds__(256)
k_agree(const float* __restrict__ uhat, const float* __restrict__ v,
        float* __restrict__ blog) {
    __shared__ float red[8];
    const int i = blockIdx.x, j = blockIdx.y, tid = threadIdx.x;
    const float* up = uhat + ((size_t)(i * OUT_NODES + j) << 12);
    const float* vp = v + j * 4096;
    float acc = 0.f;
#pragma unroll
    for (int q = 0; q < 16; ++q) {
        const int e = tid + q * 256;
        acc += up[e] * vp[e];
    }
#pragma unroll
    for (int m = 16; m > 0; m >>= 1) acc += __shfl_xor(acc, m, 32);
    if ((tid & 31) == 0) red[tid >> 5] = acc;
    __syncthreads();
    if (tid == 0) {
        float t = 0.f;
#pragma unroll
        for (int wv = 0; wv < 8; ++wv) t += red[wv];
        blog[i * OUT_NODES + j] += t * (1.0f / BATCH);
    }
}

// out[b, j, o, 1] = v[j, b, o]
__global__ void k_out(const float* __restrict__ v, float* __restrict__ out) {
    const int n = blockIdx.x * 256 + threadIdx.x;
    if (n >= BATCH * OUT_NODES * DIM) return;
    const int o  = n & 63;
    const int j  = (n >> 6) % OUT_NODES;
    const int bb = n / (OUT_NODES * 64);
    out[n] = v[(j * BATCH + bb) * 64 + o];
}

// ---------------------------------------------------------------------------
extern "C" void kernel_launch(void* const* d_in, const int* in_sizes, int n_in,
                              void* d_out, int out_size, void* d_ws, size_t ws_size,
                              hipStream_t stream) {
    const float* x = (const float*)d_in[0];   // [64, 64, 1152]
    const float* w = (const float*)d_in[1];   // [1152, 10, 64, 64]
    float* out = (float*)d_out;               // [64, 10, 64, 1]

    float* xT   = (float*)d_ws;                                  // 1152*64*64
    float* uhat = xT   + (size_t)IN_NODES * 4096;                // 1152*10*4096
    float* blog = uhat + (size_t)IN_NODES * OUT_NODES * 4096;    // 1152*10
    float* c    = blog + IN_NODES * OUT_NODES;
    float* s    = c    + IN_NODES * OUT_NODES;                   // 10*64*64
    float* v    = s    + OUT_NODES * BATCH * DIM;

    k_transpose_x<<<dim3(36, 2, 64), dim3(32, 8), 0, stream>>>(x, xT);
    k_uhat<<<IN_NODES, 256, 0, stream>>>(xT, w, uhat);
    k_zero<<<(IN_NODES * OUT_NODES + 255) / 256, 256, 0, stream>>>(blog,
                                                                   IN_NODES * OUT_NODES);
    for (int t = 0; t < ROUTING; ++t) {
        k_softmax<<<(IN_NODES + 255) / 256, 256, 0, stream>>>(blog, c);
        k_wsum<<<dim3(16, OUT_NODES), 256, 0, stream>>>(uhat, c, s);
        k_squash<<<OUT_NODES * BATCH, 32, 0, stream>>>(s, v);
        if (t < ROUTING - 1)
            k_agree<<<dim3(IN_NODES, OUT_NODES), 256, 0, stream>>>(uhat, v, blog);
    }
    k_out<<<(BATCH * OUT_NODES * DIM + 255) / 256, 256, 0, stream>>>(v, out);
}